// SkeletalEncBlock_6030134083782
// MI455X (gfx1250) — compile-verified
//
#include <hip/hip_runtime.h>

typedef float v2f __attribute__((ext_vector_type(2)));
typedef float v8f __attribute__((ext_vector_type(8)));

#define J 24
#define CIN 8
#define COUT 8
#define KW 3
#define T_FRAMES 4096
#define NCH_IN (J*CIN)     /* 192 */
#define NPOOL (J/2)        /* 12  */
#define TCHUNK 64
#define CHS 194            /* transposed LDS: row = t-col, 194 floats (even, 194%64==2) */
#define NCOL 66            /* cols 0..65  <->  t0-1 .. t0+64 */
#define NEG_SLOPE 0.2f
#define KTOT 96            /* 4 input joints * 8 ci * 3 taps */
#define NFRAG (KTOT/4)     /* 24 WMMA k-slices of 4 */

__global__ __launch_bounds__(384) void skeletal_enc_kernel(
    const float* __restrict__ x, const float* __restrict__ offset,
    const float* __restrict__ W, const float* __restrict__ bias,
    const float* __restrict__ W_off, float* __restrict__ out)
{
  __shared__ __align__(16) float lds[NCOL * CHS];   // 51216 B, layout [t_col][ch]

  const int b  = blockIdx.y;
  const int t0 = blockIdx.x * TCHUNK;
  const int tid = threadIdx.x;

  const float* xb = x + (size_t)b * NCH_IN * T_FRAMES;

  // ---- staging (transposed) via CDNA5 async DMA: x[ch, t0+tt] -> lds[(tt+1)*CHS + ch]
  // per-lane global addr + per-lane LDS addr: the async unit does the transpose
  // scatter directly, no VGPR round-trip. Tracked by ASYNCcnt.
  {
    int i = tid;
    #pragma unroll
    for (int jj = 0; jj < 32; ++jj, i += 384) {   // 192*64 elements / 384 threads
      int ch = i >> 6;                            // lanes walk consecutive tt ->
      int tt = i & 63;                            // 128B-coalesced global segments
      unsigned gvoff = (unsigned)((ch * T_FRAMES + t0 + tt) * 4);
      unsigned laddr = (unsigned)(((tt + 1) * CHS + ch) * 4);
      asm volatile("global_load_async_to_lds_b32 %0, %1, %2"
                   :: "v"(laddr), "v"(gvoff), "s"(xb) : "memory");
    }
  }
  // ---- halo cols: t0-1 -> col 0, t0+64 -> col 65 (zero at sequence edges) ----
  {
    int ch   = tid >> 1;
    int side = tid & 1;
    int tg   = side ? (t0 + TCHUNK) : (t0 - 1);
    int tc   = tg < 0 ? 0 : (tg > T_FRAMES - 1 ? T_FRAMES - 1 : tg);
    float v  = xb[ch * T_FRAMES + tc];
    if (tg < 0 || tg >= T_FRAMES) v = 0.f;        // branchless zero at edges
    lds[(side ? (NCOL - 1) : 0) * CHS + ch] = v;
  }
  asm volatile("s_wait_asynccnt 0x0" ::: "memory");
  __syncthreads();

  const int wave = tid >> 5;          // 0..11 -> joint-pair (pool region)
  const int lane = tid & 31;
  const int p = wave;
  const int n = lane & 15;            // B/C/D column
  const int h = lane >> 4;            // k-group (ISA A/B layout)
  const int m = lane & 15;            // A row

  // ---- preload masked A fragments + per-lane LDS base offsets for B ----
  // contraction order kk = jidx*24 + tap*8 + ci  (ci fastest -> element pair
  // (ci, ci+1) of the same tap is CONTIGUOUS in the transposed LDS -> ds_load_b64)
  v2f a[NFRAG];
  int ib[NFRAG];
  const int jo = 2 * p + ((m >> 3) & 1);
  const int co = m & 7;
  #pragma unroll
  for (int q = 0; q < NFRAG; ++q) {
    #pragma unroll
    for (int e = 0; e < 2; ++e) {
      int kk   = 4 * q + 2 * h + e;   // contraction index 0..95
      int jidx = kk / 24;             // candidate input joint slot (0..3)
      int r    = kk - jidx * 24;
      int tap  = r >> 3;              // 0..2
      int ci   = r & 7;               // e==0 -> even ci, e==1 -> ci+1 (same tap)
      int jin  = 2 * p - 1 + jidx;    // candidate input joint (may be out of range)
      int jc   = jin < 0 ? 0 : (jin > J - 1 ? J - 1 : jin);
      // A element: branchless masked load (clamped-safe index + cndmask)
      bool valid = (jin >= 0) & (jin < J) & (jin >= jo - 1) & (jin <= jo + 1);
      float wv = W[((jo * COUT + co) * NCH_IN + (jc * CIN + ci)) * KW + tap];
      float av = valid ? wv : 0.f;
      if (e == 0) a[q].x = av; else a[q].y = av;
      // B base (element pair start, tile col 0): col = n + tap, channel = jc*8+ci (even)
      if (e == 0) ib[q] = (n + tap) * CHS + (jc * CIN + ci);   // +tc*CHS via imm offset
    }
  }

  // ---- per-(pool,channel) constant: 0.5*(bias + masked offset-injection) ----
  float cp[8];
  const float* offb = offset + b * (J * 3);
  #pragma unroll
  for (int d = 0; d < 8; ++d) {
    float s = 0.f;
    #pragma unroll
    for (int u = 0; u < 2; ++u) {
      int joo = 2 * p + u;
      int ch  = joo * COUT + d;
      float acc = bias[ch];
      #pragma unroll
      for (int dj = -1; dj <= 1; ++dj) {
        int jin = joo + dj;
        bool valid = (jin >= 0) & (jin < J);
        int jc = jin < 0 ? 0 : (jin > J - 1 ? J - 1 : jin);
        #pragma unroll
        for (int c3 = 0; c3 < 3; ++c3) {
          float wv = W_off[ch * (J * 3) + jc * 3 + c3];
          acc += offb[jc * 3 + c3] * (valid ? wv : 0.f);
        }
      }
      s += acc;
    }
    cp[d] = 0.5f * s;
  }

  float* outp = out + ((size_t)b * (NPOOL * COUT) + p * COUT) * T_FRAMES + t0;

  // ---- main loop: four 16x16x96 GEMM tiles (64 frames), A reused 4x per frag ----
  v8f acc0 = {}, acc1 = {}, acc2 = {}, acc3 = {};
  #pragma unroll
  for (int q = 0; q < NFRAG; ++q) {
    // one ds_load_b64 per tile (compiler pairs them into ds_load_2addr_b64)
    v2f bf0 = *(const v2f*)&lds[ib[q]           ];
    v2f bf1 = *(const v2f*)&lds[ib[q] + 16 * CHS];
    v2f bf2 = *(const v2f*)&lds[ib[q] + 32 * CHS];
    v2f bf3 = *(const v2f*)&lds[ib[q] + 48 * CHS];
    acc0 = __builtin_amdgcn_wmma_f32_16x16x4_f32(false, a[q], false, bf0, (short)0, acc0, false, false);
    acc1 = __builtin_amdgcn_wmma_f32_16x16x4_f32(false, a[q], false, bf1, (short)0, acc1, false, false);
    acc2 = __builtin_amdgcn_wmma_f32_16x16x4_f32(false, a[q], false, bf2, (short)0, acc2, false, false);
    acc3 = __builtin_amdgcn_wmma_f32_16x16x4_f32(false, a[q], false, bf3, (short)0, acc3, false, false);
  }

  // ---- pooling (rows d and d+8 share a VGPR), epilogue, stores ----
  #pragma unroll
  for (int d = 0; d < 8; ++d) {
    float s0 = acc0[d] + __shfl_xor(acc0[d], 16, 32);
    float s1 = acc1[d] + __shfl_xor(acc1[d], 16, 32);
    float s2 = acc2[d] + __shfl_xor(acc2[d], 16, 32);
    float s3 = acc3[d] + __shfl_xor(acc3[d], 16, 32);
    float cA = (lane < 16) ? s0 : s1;            // frames t0+0  .. t0+31
    float cB = (lane < 16) ? s2 : s3;            // frames t0+32 .. t0+63
    float rA = 0.5f * cA + cp[d];
    float rB = 0.5f * cB + cp[d];
    rA = rA > 0.f ? rA : NEG_SLOPE * rA;         // LeakyReLU(0.2)
    rB = rB > 0.f ? rB : NEG_SLOPE * rB;
    outp[d * T_FRAMES + lane]      = rA;         // 128B contiguous stores
    outp[d * T_FRAMES + 32 + lane] = rB;
  }
}

extern "C" void kernel_launch(void* const* d_in, const int* in_sizes, int n_in,
                              void* d_out, int out_size, void* d_ws, size_t ws_size,
                              hipStream_t stream) {
  const float* x      = (const float*)d_in[0];
  const float* offset = (const float*)d_in[1];
  const float* W      = (const float*)d_in[2];
  const float* bias   = (const float*)d_in[3];
  const float* W_off  = (const float*)d_in[4];
  float* out = (float*)d_out;
  const int Bv = in_sizes[0] / (NCH_IN * T_FRAMES);   // 128
  dim3 grid(T_FRAMES / TCHUNK, Bv);
  skeletal_enc_kernel<<<grid, 384, 0, stream>>>(x, offset, W, bias, W_off, out);
}